// BehaviorFluidFlow_30296699306432
// MI455X (gfx1250) — compile-verified
//
#include <hip/hip_runtime.h>
#include <hip/hip_bf16.h>
#include <stdint.h>

// ---- CDNA5 async global->LDS availability probe (guarded; falls back) ----
#if defined(__has_builtin)
#if __has_builtin(__builtin_amdgcn_global_load_async_to_lds_b32) && \
    __has_builtin(__builtin_amdgcn_s_wait_asynccnt)
#define ASYNC_OK 1
#endif
#endif
#ifndef ASYNC_OK
#define ASYNC_OK 0
#endif

namespace {
constexpr int kW   = 512;
constexpr int kHW  = 512 * 512;
constexpr int kC   = 20;
constexpr int kCHW = kC * kHW;
constexpr int kB   = 16;

// element-id bitmasks (ids are exact small integers stored as f32)
constexpr uint32_t kAirMask  = (1u << 14) | (1u << 15);               // agentKangaroo|agentLemming
constexpr uint32_t kElemMask = (1u << 0) | (1u << 3) | (1u << 9) |
                               (1u << 8) | (1u << 12) | kAirMask;      // empty|water|gas|lava|acid|air
// final is_fluid set is identical to kElemMask: {empty,water,gas,lava,acid,agentK,agentL}
}  // namespace

#if ASYNC_OK
typedef __attribute__((address_space(1))) int global_int_t;
typedef __attribute__((address_space(3))) int lds_int_t;
#endif

__device__ __forceinline__ void row_load(const float* gp, float* lp) {
#if ASYNC_OK
  __builtin_amdgcn_global_load_async_to_lds_b32(
      (global_int_t*)const_cast<float*>(gp),
      (lds_int_t*)lp,
      /*offset=*/0, /*cpol=*/0);
#else
  *lp = *gp;
#endif
}

__device__ __forceinline__ void row_load_wait() {
#if ASYNC_OK
  __builtin_amdgcn_s_wait_asynccnt(0);
#endif
}

__global__ __launch_bounds__(kW)
void fluid_flow_kernel(const float* __restrict__ world,
                       const float* __restrict__ rand_mov,
                       float* __restrict__ out) {
  __shared__ float s_id[kW], s_dens[kW], s_grav[kW], s_mom[kW], s_didg[kW], s_rand[kW];
  __shared__ int   s_fA[kW], s_fB[kW], s_perm[kW];

  const int tid  = threadIdx.x;
  const int row  = blockIdx.x;       // 0 .. B*H-1
  const int b    = row >> 9;
  const int h    = row & 511;
  const int base  = b * kCHW + h * kW;   // + c*kHW later; max ~84M fits int32
  const int rbase = b * kHW + h * kW;

  // Stage the 5 decision channels + rand into LDS (async on CDNA5, ASYNCcnt-tracked).
  row_load(world + base + 0 * kHW + tid, &s_id[tid]);
  row_load(world + base + 1 * kHW + tid, &s_dens[tid]);
  row_load(world + base + 2 * kHW + tid, &s_grav[tid]);
  row_load(world + base + 6 * kHW + tid, &s_mom[tid]);
  row_load(world + base + 8 * kHW + tid, &s_didg[tid]);
  row_load(rand_mov + rbase + tid, &s_rand[tid]);
  row_load_wait();
  __syncthreads();

  const int wm1 = (tid - 1) & (kW - 1);
  const int wp1 = (tid + 1) & (kW - 1);

  // ---------------- pass 1: fall left (neighbor = w-1) ----------------
  {
    const int   iid  = (int)s_id[tid];
    const bool  air  = (kAirMask >> iid) & 1u;
    const bool  elem = (kElemMask >> iid) & 1u;
    const bool  fd   = ((s_rand[tid] + s_mom[tid]) + 0.0f) > 0.5f;  // nfm == 0
    const bool  ndg  = (!(s_didg[tid] > 0.0f)) || air;
    const bool  dl   = (s_dens[tid] - s_dens[wm1]) > 0.0f;
    const bool  g    = (s_grav[tid] == 1.0f);
    const bool  gl   = (s_grav[wm1] == 1.0f);
    s_fA[tid] = (fd && elem && ndg && dl && gl && g) ? 1 : 0;
  }
  __syncthreads();
  const int dbl1   = s_fA[tid];
  const int dbl1n  = s_fA[wp1];
  const int d1r    = dbl1 & (dbl1n ^ 1);        // does_become_left_real
  s_fB[tid] = d1r;
  __syncthreads();
  const int   dbrr1 = s_fB[wp1];                // does_become_right_real
  const float nfm1  = 2.0f * (float)dbrr1;      // momentum after pass 1
  const int   p1    = d1r ? wm1 : (dbrr1 ? wp1 : tid);
  s_perm[tid] = p1;
  __syncthreads();

  // ---------------- pass 2: fall right (neighbor = w+1), on permuted world ----------------
  {
    const int   p1w  = s_perm[tid];
    const int   p1n  = s_perm[wp1];
    const int   iid  = (int)s_id[p1w];
    const bool  air  = (kAirMask >> iid) & 1u;
    const bool  elem = (kElemMask >> iid) & 1u;
    const bool  fd   = ((s_rand[tid] + s_mom[p1w]) + nfm1) > 0.5f;
    const bool  ndg  = (!(s_didg[p1w] > 0.0f)) || air;
    const bool  dl   = (s_dens[p1w] - s_dens[p1n]) > 0.0f;
    const bool  g    = (s_grav[p1w] == 1.0f);
    const bool  gl   = (s_grav[p1n] == 1.0f);
    s_fA[tid] = ((!fd) && elem && ndg && dl && gl && g) ? 1 : 0;
  }
  __syncthreads();
  const int dbl2  = s_fA[tid];
  const int dbl2p = s_fA[wm1];
  const int d2r   = dbl2 & (dbl2p ^ 1);         // does_become_left_real (dir = right)
  s_fB[tid] = d2r;
  __syncthreads();
  const int   dbrr2 = s_fB[wm1];
  const float nfm   = nfm1 - 2.0f * (float)dbrr2;
  const int   q     = d2r ? wp1 : (dbrr2 ? wm1 : tid);
  const int   f     = s_perm[q];                // final source column for this output column

  // ---------------- finalize & write (all stores non-temporal: zero reuse) ----------------
  const float idf   = s_id[f];
  const int   iidf  = (int)idf;
  const bool  fluid = (kElemMask >> iidf) & 1u;
  const float out6  = fluid ? nfm : s_mom[f];

  __builtin_nontemporal_store(idf,       out + base + 0 * kHW + tid);
  __builtin_nontemporal_store(s_dens[f], out + base + 1 * kHW + tid);
  __builtin_nontemporal_store(s_grav[f], out + base + 2 * kHW + tid);
  __builtin_nontemporal_store(out6,      out + base + 6 * kHW + tid);
  __builtin_nontemporal_store(s_didg[f], out + base + 8 * kHW + tid);

  // Remaining 15 channels: stream once with the near-identity gather (|f-w|<=2 mod 512,
  // stays inside this row's 2KB line set). Single-touch data -> NT load + NT store.
  const int rest[15] = {3, 4, 5, 7, 9, 10, 11, 12, 13, 14, 15, 16, 17, 18, 19};
#pragma unroll
  for (int k = 0; k < 15; ++k) {
    const int c = rest[k];
    const float v = __builtin_nontemporal_load(world + base + c * kHW + f);
    __builtin_nontemporal_store(v, out + base + c * kHW + tid);
  }
}

extern "C" void kernel_launch(void* const* d_in, const int* in_sizes, int n_in,
                              void* d_out, int out_size, void* d_ws, size_t ws_size,
                              hipStream_t stream) {
  (void)in_sizes; (void)n_in; (void)d_ws; (void)ws_size; (void)out_size;
  const float* world = (const float*)d_in[0];
  const float* rmov  = (const float*)d_in[1];  // rand_interact / rand_element unused by reference
  float* out = (float*)d_out;
  fluid_flow_kernel<<<dim3(kB * 512), dim3(kW), 0, stream>>>(world, rmov, out);
}